// Cross_attention_3_71159018160211
// MI455X (gfx1250) — compile-verified
//
#include <hip/hip_runtime.h>
#include <stdint.h>

// ---- problem constants (from reference) ----
#define CCH 64                 // channels (in == out)
#define DD  72
#define HH  108
#define WW  108
#define HWD (HH*WW)            // 11664
#define NVOX (DD*HWD)          // 839808 voxels
#define PP  9
#define P2  81                 // 9*9 patch
#define HB  (HWD/PP)           // 1296 patches along HW
#define LL  ((DD/PP)*HB)       // 10368 patches total
// padded GEMM dims
#define K1P 84                 // 81 -> 84  (K of GEMM1, att GEMM)
#define N1  162
#define N1P 176                // 162 -> 176 (N of GEMM1)
#define K2P 164                // 162 -> 164 (K of GEMM2)
#define N2P 96                 // 81 -> 96   (N of GEMM2, att GEMM)

typedef float v2f __attribute__((ext_vector_type(2)));
typedef float v8f __attribute__((ext_vector_type(8)));
typedef uint32_t u32x4 __attribute__((ext_vector_type(4)));
typedef uint32_t u32x8 __attribute__((ext_vector_type(8)));

// async-to-LDS builtin wants int* params (per hipcc diagnostic round 2)
#define AS1I(p) ((__attribute__((address_space(1))) int*)(p))
#define AS3I(p) ((__attribute__((address_space(3))) int*)(p))

static __device__ __forceinline__ uint32_t lds_offset(const void* p) {
    return (uint32_t)(size_t)(__attribute__((address_space(3))) const void*)p;
}

static __device__ __forceinline__ void wait_tensor0() {
#if __has_builtin(__builtin_amdgcn_s_wait_tensorcnt)
    __builtin_amdgcn_s_wait_tensorcnt(0);
#else
    asm volatile("s_wait_tensorcnt 0x0" ::: "memory");
#endif
}

#if __has_builtin(__builtin_amdgcn_global_load_async_to_lds_b32)
#define HAVE_ASYNC_LDS 1
static __device__ __forceinline__ void wait_async0() {
#if __has_builtin(__builtin_amdgcn_s_wait_asynccnt)
    __builtin_amdgcn_s_wait_asynccnt(0);
#else
    asm volatile("s_wait_asynccnt 0x0" ::: "memory");
#endif
}
#else
#define HAVE_ASYNC_LDS 0
#endif

// D = A(16x4) * B(4x16) + C, fp32 exact — V_WMMA_F32_16X16X4_F32
static __device__ __forceinline__ v8f wmma4(v2f a, v2f b, v8f c) {
    return __builtin_amdgcn_wmma_f32_16x16x4_f32(
        /*neg_a=*/false, a, /*neg_b=*/false, b,
        /*c_mod=*/(short)0, c, /*reuse_a=*/false, /*reuse_b=*/false);
}

// TDM: DMA a 2-D tile (tile_d0 elems contiguous x tile_d1 rows, row stride
// `stride` elems, fp32) from global into LDS. Wave-level op; EXEC ignored.
// Descriptor per CDNA5 ISA "8.3 D# Group 0" / "8.4 D# Group 1".
static __device__ __forceinline__ void tdm_load_tile_f32(
        const float* gsrc, uint32_t lds_bytes,
        uint32_t tile_d0, uint32_t tile_d1, uint32_t stride_elems) {
    uint64_t ga = (uint64_t)(uintptr_t)gsrc;
    u32x4 g0;
    g0[0] = 0x1u;                                   // count=1, user mode
    g0[1] = lds_bytes;                              // lds_addr
    g0[2] = (uint32_t)ga;                           // global_addr[31:0]
    g0[3] = (uint32_t)(ga >> 32) | (2u << 30);      // global_addr[56:32] | type=2
    u32x8 g1;
    g1[0] = 2u << 16;                               // data_size=2 (4 bytes)
    g1[1] = (stride_elems & 0xFFFFu) << 16;         // tensor_dim0[15:0]  (use stride as extent)
    g1[2] = (stride_elems >> 16) | (tile_d1 << 16); // tensor_dim0[31:16] | tensor_dim1[15:0]
    g1[3] = (tile_d0 << 16);                        // tensor_dim1[31:16]=0 | tile_dim0
    g1[4] = tile_d1;                                // tile_dim1 | tile_dim2=0
    g1[5] = stride_elems;                           // tensor_dim0_stride[31:0]
    g1[6] = 0u;                                     // stride hi | dim1_stride lo
    g1[7] = 0u;
    asm volatile("tensor_load_to_lds %0, %1" :: "s"(g0), "s"(g1) : "memory");
}

// ---- pack w1 (162,81) -> w1p (84x176) = zero-padded w1^T ;
//      pack w2 (81,162) -> w2p (164x96) = zero-padded w2^T ----
__global__ void pack_weights(const float* __restrict__ w1, const float* __restrict__ w2,
                             float* __restrict__ w1p, float* __restrict__ w2p) {
    const int tot1 = K1P * N1P;
    const int tot2 = K2P * N2P;
    for (int idx = blockIdx.x * blockDim.x + threadIdx.x; idx < tot1 + tot2;
         idx += gridDim.x * blockDim.x) {
        if (idx < tot1) {
            int k = idx / N1P, n = idx % N1P;
            w1p[idx] = (k < P2 && n < N1) ? w1[n * P2 + k] : 0.0f;
        } else {
            int j = idx - tot1;
            int k = j / N2P, n = j % N2P;
            w2p[j] = (k < N1 && n < P2) ? w2[n * N1 + k] : 0.0f;
        }
    }
}

// ---- 1x1x1 conv projections: out[o,v] = sum_c W[o,c]*in[c,v] + b[o]
// block: 128 thr (4 waves), wave w owns M-tile rows 16w..16w+15; gridDim.y picks x/y.
// Input tile (64 ch x 16 voxels) DMA'd to LDS by the Tensor Data Mover.
__global__ void __launch_bounds__(128) conv1x1(
        const float* __restrict__ x, const float* __restrict__ y,
        const float* __restrict__ w_img, const float* __restrict__ b_img,
        const float* __restrict__ w_fea, const float* __restrict__ b_fea,
        float* __restrict__ xc, float* __restrict__ yc) {
    const int n0 = blockIdx.x * 16;
    const bool second = (blockIdx.y != 0);
    const float* src = second ? y : x;
    const float* w   = second ? w_fea : w_img;
    const float* bia = second ? b_fea : b_img;
    float* dst       = second ? yc : xc;

    __shared__ float Xs[64][16];               // K x N tile, shared by all 4 waves
    if ((threadIdx.x >> 5) == 0) {             // one wave issues the TDM DMA
        tdm_load_tile_f32(src + n0, lds_offset(&Xs[0][0]),
                          /*tile_d0=*/16, /*tile_d1=*/64, /*stride=*/NVOX);
        wait_tensor0();
    }
    __syncthreads();

    const int lane = threadIdx.x & 31;
    const int wv   = threadIdx.x >> 5;         // M-tile id
    const int m0   = wv * 16;
    const int lo   = lane & 15;
    const int hi   = lane >> 4;

    v8f acc = {};
    for (int k0 = 0; k0 < 64; k0 += 4) {
        v2f a, b;
        // A 16x4 layout: lane row = lo, K = k0 + vgpr + 2*hi
        a.x = w[(m0 + lo) * 64 + k0 + 2 * hi + 0];
        a.y = w[(m0 + lo) * 64 + k0 + 2 * hi + 1];
        // B 4x16 layout (mirror): col = lo, K = k0 + vgpr + 2*hi
        b.x = Xs[k0 + 2 * hi + 0][lo];
        b.y = Xs[k0 + 2 * hi + 1][lo];
        acc = wmma4(a, b, acc);
    }
#pragma unroll
    for (int v = 0; v < 8; ++v) {
        int row = m0 + v + 8 * hi;             // C/D layout: M = vgpr + 8*(lane/16)
        dst[(size_t)row * NVOX + n0 + lo] = acc[v] + bia[row];
    }
}

// ---- unfold + MLP (81->162->81, LeakyReLU 0.2) for one channel / 16 L-rows ----
// writes packed (C, L, 81) activations; gridDim.z picks img/fea path.
__global__ void __launch_bounds__(128) unfold_mlp(
        const float* __restrict__ xc, const float* __restrict__ yc,
        const float* __restrict__ w1p, const float* __restrict__ w2p,
        float* __restrict__ img, float* __restrict__ fea) {
    const int  l0     = blockIdx.x * 16;
    const int  c      = blockIdx.y;
    const bool second = (blockIdx.z != 0);
    const float* src  = second ? yc : xc;
    float* dst        = second ? fea : img;

    __shared__ float As[16][K1P];              // unfolded patch rows, K-padded
    __shared__ float Ts[16][N1P];              // intermediate (Linear1 out), padded

    for (int i = threadIdx.x; i < 16 * K1P; i += 128) {
        int r = i / K1P, k = i % K1P;
        float v = 0.0f;
        if (k < P2) {
            int l = l0 + r;
            int dblk = l / HB, hwblk = l % HB;
            int pr = k / PP, pc = k % PP;
            v = src[(size_t)c * NVOX + (size_t)(dblk * PP + pr) * HWD + hwblk * PP + pc];
        }
        As[r][k] = v;
    }
    __syncthreads();

    const int lane = threadIdx.x & 31;
    const int wv   = threadIdx.x >> 5;
    const int lo   = lane & 15, hi = lane >> 4;

    // GEMM1: (16 x 84) @ (84 x 176) -> Ts ; padded cols come out exactly 0
    for (int nt = wv; nt < N1P / 16; nt += 4) {
        v8f acc = {};
        for (int k0 = 0; k0 < K1P; k0 += 4) {
            v2f a, b;
            a.x = As[lo][k0 + 2 * hi + 0];
            a.y = As[lo][k0 + 2 * hi + 1];
            b.x = w1p[(k0 + 2 * hi + 0) * N1P + nt * 16 + lo];
            b.y = w1p[(k0 + 2 * hi + 1) * N1P + nt * 16 + lo];
            acc = wmma4(a, b, acc);
        }
#pragma unroll
        for (int v = 0; v < 8; ++v)
            Ts[v + 8 * hi][nt * 16 + lo] = acc[v];
    }
    __syncthreads();

    // GEMM2: (16 x 164) @ (164 x 96) -> LeakyReLU -> packed (C,L,81)
    for (int nt = wv; nt < N2P / 16; nt += 4) {
        v8f acc = {};
        for (int k0 = 0; k0 < K2P; k0 += 4) {
            v2f a, b;
            a.x = Ts[lo][k0 + 2 * hi + 0];
            a.y = Ts[lo][k0 + 2 * hi + 1];
            b.x = w2p[(k0 + 2 * hi + 0) * N2P + nt * 16 + lo];
            b.y = w2p[(k0 + 2 * hi + 1) * N2P + nt * 16 + lo];
            acc = wmma4(a, b, acc);
        }
        int col = nt * 16 + lo;
        if (col < P2) {
#pragma unroll
            for (int v = 0; v < 8; ++v) {
                float val = acc[v];
                val = (val > 0.0f) ? val : 0.2f * val;   // LeakyReLU(0.2)
                dst[((size_t)c * LL + l0 + v + 8 * hi) * P2 + col] = val;
            }
        }
    }
}

// ---- AdaptiveAvgPool1d(81): pooled[c,p,q] = mean_{r<128} fea[c, q*128+r, p]
// stored zero-padded as B matrix (84 x 96) per channel for the att GEMM.
__global__ void __launch_bounds__(96) pool81(const float* __restrict__ fea,
                                             float* __restrict__ pooledp) {
    const int q = blockIdx.x;                  // 0..80  (N index of att GEMM)
    const int c = blockIdx.y;
    const int p = threadIdx.x;                 // 0..95  (K index)
    if (p >= P2) return;
    float acc = 0.0f;
    const float* base = fea + ((size_t)c * LL + (size_t)q * 128) * P2 + p;
    for (int r = 0; r < 128; ++r) acc += base[(size_t)r * P2];
    pooledp[(size_t)c * K1P * N2P + p * N2P + q] = acc * (1.0f / 128.0f);
}

// ---- att = img @ pooled per channel, folded straight to (C,D,H,W) output ----
// block: 192 thr (6 waves); wave = one 16-col N-tile of the 96-wide padded output.
// A tile staged cache->LDS with async loads (ASYNCcnt) when available.
__global__ void __launch_bounds__(192) att_fold(const float* __restrict__ img,
                                                const float* __restrict__ pooledp,
                                                float* __restrict__ out) {
    const int l0 = blockIdx.x * 16;
    const int c  = blockIdx.y;

    __shared__ float As[16][K1P];
#if HAVE_ASYNC_LDS
    for (int i = threadIdx.x; i < 16 * K1P; i += 192) {
        int r = i / K1P, k = i % K1P;
        if (k < P2) {
            __builtin_amdgcn_global_load_async_to_lds_b32(
                AS1I(img + ((size_t)c * LL + l0 + r) * P2 + k),
                AS3I(&As[r][k]), 0, 0);
        } else {
            As[r][k] = 0.0f;
        }
    }
    wait_async0();
#else
    for (int i = threadIdx.x; i < 16 * K1P; i += 192) {
        int r = i / K1P, k = i % K1P;
        As[r][k] = (k < P2) ? img[((size_t)c * LL + l0 + r) * P2 + k] : 0.0f;
    }
#endif
    __syncthreads();

    const int lane = threadIdx.x & 31;
    const int nt   = threadIdx.x >> 5;         // 0..5
    const int lo   = lane & 15, hi = lane >> 4;
    const float* B = pooledp + (size_t)c * K1P * N2P;

    v8f acc = {};
    for (int k0 = 0; k0 < K1P; k0 += 4) {
        v2f a, b;
        a.x = As[lo][k0 + 2 * hi + 0];
        a.y = As[lo][k0 + 2 * hi + 1];
        b.x = B[(k0 + 2 * hi + 0) * N2P + nt * 16 + lo];
        b.y = B[(k0 + 2 * hi + 1) * N2P + nt * 16 + lo];
        acc = wmma4(a, b, acc);
    }

    int m = nt * 16 + lo;
    if (m < P2) {
        int pr = m / PP, pc = m % PP;
#pragma unroll
        for (int v = 0; v < 8; ++v) {
            int l = l0 + v + 8 * hi;
            int dblk = l / HB, hwblk = l % HB;
            out[(size_t)c * NVOX + (size_t)(dblk * PP + pr) * HWD + hwblk * PP + pc] = acc[v];
        }
    }
}

extern "C" void kernel_launch(void* const* d_in, const int* in_sizes, int n_in,
                              void* d_out, int out_size, void* d_ws, size_t ws_size,
                              hipStream_t stream) {
    const float* x     = (const float*)d_in[0];
    const float* y     = (const float*)d_in[1];
    const float* w_img = (const float*)d_in[2];
    const float* b_img = (const float*)d_in[3];
    const float* w_fea = (const float*)d_in[4];
    const float* b_fea = (const float*)d_in[5];
    const float* w1    = (const float*)d_in[6];
    const float* w2    = (const float*)d_in[7];
    float* out = (float*)d_out;

    float* ws = (float*)d_ws;
    size_t off = 0;
    float* xc      = ws + off; off += (size_t)CCH * NVOX;      // 215 MB
    float* yc      = ws + off; off += (size_t)CCH * NVOX;      // 215 MB
    float* img     = ws + off; off += (size_t)CCH * LL * P2;   // 215 MB
    float* fea     = ws + off; off += (size_t)CCH * LL * P2;   // 215 MB
    float* w1p     = ws + off; off += (size_t)K1P * N1P;
    float* w2p     = ws + off; off += (size_t)K2P * N2P;
    float* pooledp = ws + off; off += (size_t)CCH * K1P * N2P; // 2 MB

    pack_weights<<<64, 256, 0, stream>>>(w1, w2, w1p, w2p);
    conv1x1<<<dim3(NVOX / 16, 2), 128, 0, stream>>>(x, y, w_img, b_img, w_fea, b_fea, xc, yc);
    unfold_mlp<<<dim3(LL / 16, CCH, 2), 128, 0, stream>>>(xc, yc, w1p, w2p, img, fea);
    (void)hipMemsetAsync(pooledp, 0, (size_t)CCH * K1P * N2P * sizeof(float), stream);
    pool81<<<dim3(P2, CCH), 96, 0, stream>>>(fea, pooledp);
    att_fold<<<dim3(LL / 16, CCH), 192, 0, stream>>>(img, pooledp, out);
}